// MultiHeadAttention_42064909697067
// MI455X (gfx1250) — compile-verified
//
#include <hip/hip_runtime.h>
#include <hip/hip_bf16.h>
#include <math.h>
#include <float.h>
#include <stdint.h>

typedef __attribute__((ext_vector_type(2))) float v2f;
typedef __attribute__((ext_vector_type(8))) float v8f;

#define B_  2
#define L_  2048
#define D_  768
#define H_  12
#define HD_ 64
#define N3_ (3 * D_)
#define KPAD_ 68   // padded LDS row stride (floats) for staged K tiles: 272B,
                   // 16B-aligned for b128 async copies, bank-conflict-free reads

static __device__ __forceinline__ v8f wmma_f32(v2f a, v2f b, v8f c) {
    // V_WMMA_F32_16X16X4_F32 : D(16x16 f32) = A(16x4 f32) * B(4x16 f32) + C
    return __builtin_amdgcn_wmma_f32_16x16x4_f32(
        /*neg_a=*/false, a, /*neg_b=*/false, b,
        /*c_mod=*/(short)0, c, /*reuse_a=*/false, /*reuse_b=*/false);
}

// ---------------------------------------------------------------------------
// Kernel 1: QKV projection + bias + RoPE, scatter to [B,H,L,HD] Q/K/V buffers.
// One wave computes a 16(M) x 64(N) block (full head width) so rotate_half's
// partner column (d +/- 32) is in the same lane two accumulators away.
// ---------------------------------------------------------------------------
__global__ void qkv_rope_kernel(const float* __restrict__ X,
                                const float* __restrict__ W,
                                const float* __restrict__ bias,
                                float* __restrict__ Qo,
                                float* __restrict__ Ko,
                                float* __restrict__ Vo) {
    const int nb   = blockIdx.x;                 // 0..35 : 64-wide N block of qkv
    const int wave = threadIdx.x >> 5;
    const int mt   = blockIdx.y * 4 + wave;      // 0..255 : 16-row M tile
    const int lane = threadIdx.x & 31;
    const int col  = lane & 15;
    const int half = lane >> 4;
    const int m0   = mt * 16;
    const int nbase = nb * 64;

    const float* arow = X + (size_t)(m0 + col) * D_;

    v8f c0 = {}, c1 = {}, c2 = {}, c3 = {};
    for (int k = 0; k < D_; k += 4) {
        // A fragment: lanes0-15 hold row=lane K=(k,k+1); lanes16-31 K=(k+2,k+3)
        v2f a = *(const v2f*)(arow + k + half * 2);
        const float* w0 = W + (size_t)(k + half * 2)     * N3_ + nbase;
        const float* w1 = W + (size_t)(k + half * 2 + 1) * N3_ + nbase;
        v2f b;
        b.x = w0[col +  0]; b.y = w1[col +  0]; c0 = wmma_f32(a, b, c0);
        b.x = w0[col + 16]; b.y = w1[col + 16]; c1 = wmma_f32(a, b, c1);
        b.x = w0[col + 32]; b.y = w1[col + 32]; c2 = wmma_f32(a, b, c2);
        b.x = w0[col + 48]; b.y = w1[col + 48]; c3 = wmma_f32(a, b, c3);
    }
    v8f acc[4] = {c0, c1, c2, c3};
    for (int t = 0; t < 4; ++t) {
        float bv = bias[nbase + t * 16 + col];
        for (int i = 0; i < 8; ++i) acc[t][i] += bv;
    }

    const int bat = m0 >> 11;          // batch (tiles never straddle)
    const int l0  = m0 & (L_ - 1);

    float* dstbase;
    int h;
    bool rope;
    if      (nb < 12) { dstbase = Qo; h = nb;      rope = true;  }
    else if (nb < 24) { dstbase = Ko; h = nb - 12; rope = true;  }
    else              { dstbase = Vo; h = nb - 24; rope = false; }

    v8f outv[4];
    if (rope) {
        for (int t = 0; t < 4; ++t) {
            const int   fi  = (t & 1) * 16 + col;                // freq index d % 32
            const float inv = __expf(-(float)fi * 0.28782313662425572f); // ln(1e4)/32
            const float sgn = (t < 2) ? -1.0f : 1.0f;            // -x2 | +x1
            const v8f&  prt = acc[t ^ 2];                        // partner d +/- 32
            for (int i = 0; i < 8; ++i) {
                const int   l   = l0 + i + half * 8;
                const float ang = (float)l * inv;
                outv[t][i] = acc[t][i] * cosf(ang) + sgn * prt[i] * sinf(ang);
            }
        }
    } else {
        for (int t = 0; t < 4; ++t) outv[t] = acc[t];
    }

    float* dst = dstbase + ((size_t)bat * H_ + h) * L_ * HD_;
    for (int t = 0; t < 4; ++t)
        for (int i = 0; i < 8; ++i) {
            const int l = l0 + i + half * 8;
            dst[(size_t)l * HD_ + t * 16 + col] = outv[t][i];
        }
}

// ---------------------------------------------------------------------------
// Kernel 2: fused attention. One WG (8 waves) per (b, h, 32-query block).
// Scores for the whole 32x2048 block live in LDS (256KB of the 320KB WGP LDS)
// -> probs are written to HBM exactly once and PV consumes them from LDS.
// K tiles are staged into LDS with GLOBAL_LOAD_ASYNC_TO_LDS_B128 (ASYNCcnt),
// turning the 256B-strided K^T fragment fetches into coalesced 16B/lane copies.
// ---------------------------------------------------------------------------
__global__ void attn_kernel(const float* __restrict__ Q,
                            const float* __restrict__ K,
                            const float* __restrict__ V,
                            float* __restrict__ attn_ws,
                            float* __restrict__ probs) {
    extern __shared__ float smem[];
    float* sS = smem;                         // 32 x 2048 scores/probs  (256KB)
    float* sQ = smem + 32 * L_;               // 32 x 64 staged Q          (8KB)
    float* sO = sQ + 32 * HD_;                // 32 x 64 output accum      (8KB)
    float* sKall = sO + 32 * HD_;             // 8 waves x 16 x KPAD_     (34KB)

    const int q0 = blockIdx.x * 32;
    const int h  = blockIdx.y;
    const int b  = blockIdx.z;
    const int bh = b * H_ + h;
    const float* Qp = Q + (size_t)bh * L_ * HD_;
    const float* Kp = K + (size_t)bh * L_ * HD_;
    const float* Vp = V + (size_t)bh * L_ * HD_;

    const int tid  = threadIdx.x;
    const int wave = tid >> 5;
    const int lane = tid & 31;
    const int col  = lane & 15;
    const int half = lane >> 4;

    float* sK = sKall + wave * (16 * KPAD_);           // this wave's K stage
    const uint32_t sk_off = (uint32_t)(uintptr_t)(void*)sK; // LDS byte offset

    for (int i = 0; i < 8; ++i) {        // stage Q block, zero O accumulator
        const int idx = tid * 8 + i;     // 0..2047
        sQ[idx] = Qp[(size_t)(q0 + (idx >> 6)) * HD_ + (idx & 63)];
        sO[idx] = 0.0f;
    }
    __syncthreads();

    // ---- phase 1: S = Q K^T * (1/sqrt(L)), causal mask -> LDS ----
    const float scale = 0.022097086912079608f;   // 1/sqrt(2048)
    const int kbeg = wave * 256;                  // this wave's key range
    for (int kt = 0; kt < 16; ++kt) {
        const int key0 = kbeg + kt * 16;
        const bool needed = (key0 <= q0 + 31);
        if (needed) {
            // async-stage K[key0..key0+15][0..63] -> sK (rows padded to KPAD_)
            const uint64_t gbase =
                (uint64_t)(uintptr_t)(Kp + (size_t)key0 * HD_);
            for (int j = 0; j < 8; ++j) {
                const int idx = j * 32 + lane;       // 0..255 16B chunks
                const int row = idx >> 4;            // key row 0..15
                const int chk = idx & 15;            // 16B chunk in row
                const uint32_t loff = sk_off + (uint32_t)(row * (KPAD_ * 4) + chk * 16);
                const uint64_t gaddr = gbase + (uint64_t)(row * (HD_ * 4) + chk * 16);
                asm volatile("global_load_async_to_lds_b128 %0, %1, off"
                             :: "v"(loff), "v"(gaddr) : "memory");
            }
            asm volatile("s_wait_asynccnt 0x0" ::: "memory");
        }
        for (int mtile = 0; mtile < 2; ++mtile) {
            const int mb = mtile * 16;
            if (key0 > q0 + mb + 15) {            // fully masked: skip GEMM
                for (int i = 0; i < 8; ++i)
                    sS[(size_t)(mb + i + half * 8) * L_ + key0 + col] = -FLT_MAX;
                continue;
            }
            v8f acc = {};
            const float* qrow = sQ + (mb + col) * HD_;
            const float* krow = sK + col * KPAD_;  // B[kd][key]=K[key][kd]
            for (int kk = 0; kk < HD_; kk += 4) {
                v2f a  = *(const v2f*)(qrow + kk + half * 2);
                v2f bb = *(const v2f*)(krow + kk + half * 2);
                acc = wmma_f32(a, bb, acc);
            }
            for (int i = 0; i < 8; ++i) {
                const int m   = mb + i + half * 8;
                const int key = key0 + col;
                sS[(size_t)m * L_ + key] =
                    (key > q0 + m) ? -FLT_MAX : acc[i] * scale;
            }
        }
    }
    __syncthreads();

    // ---- phase 2: row softmax (wave32 shuffle reductions), stream probs ----
    for (int i = 0; i < 4; ++i) {
        const int r  = wave * 4 + i;       // local row 0..31
        const int qg = q0 + r;             // global query index
        float* row = sS + (size_t)r * L_;
        float m = -FLT_MAX;
        for (int j = lane; j < L_; j += 32) m = fmaxf(m, row[j]);
        for (int o = 16; o > 0; o >>= 1)   m = fmaxf(m, __shfl_xor(m, o));
        float s = 0.0f;
        for (int j = lane; j < L_; j += 32) {
            const float e = __expf(row[j] - m);  // masked -> exp(-inf) = 0
            row[j] = e;
            s += e;
        }
        for (int o = 16; o > 0; o >>= 1) s += __shfl_xor(s, o);
        const float invs = 1.0f / s;
        float* prow = probs + ((size_t)bh * L_ + qg) * L_;
        for (int j = lane; j < L_; j += 32) {
            const float p = row[j] * invs;
            row[j]  = p;
            prow[j] = p;
        }
    }
    __syncthreads();

    // ---- phase 3: O = P V (skip keys with p==0), LDS-atomic cross-wave sum --
    int kmax = q0 + 32 - kbeg;             // relevant keys in this wave's slice
    if (kmax > 256) kmax = 256;
    if (kmax > 0) {
        const int chunks = (kmax + 3) >> 2;
        for (int mtile = 0; mtile < 2; ++mtile) {
            for (int nt = 0; nt < 4; ++nt) {
                v8f acc = {};
                const float* prow = sS + (size_t)(mtile * 16 + col) * L_ + kbeg;
                const float* vcol = Vp + (size_t)kbeg * HD_ + nt * 16 + col;
                for (int c = 0; c < chunks; ++c) {
                    v2f a = *(const v2f*)(prow + c * 4 + half * 2);
                    v2f bb;
                    bb.x = vcol[(size_t)(c * 4 + half * 2)     * HD_];
                    bb.y = vcol[(size_t)(c * 4 + half * 2 + 1) * HD_];
                    acc = wmma_f32(a, bb, acc);
                }
                for (int i = 0; i < 8; ++i)
                    atomicAdd(&sO[(mtile * 16 + i + half * 8) * HD_ + nt * 16 + col],
                              acc[i]);
            }
        }
    }
    __syncthreads();

    for (int i = 0; i < 8; ++i) {          // write b h l d -> b l (h d)
        const int idx = tid * 8 + i;
        attn_ws[((size_t)b * L_ + q0 + (idx >> 6)) * D_ + h * HD_ + (idx & 63)]
            = sO[idx];
    }
}

// ---------------------------------------------------------------------------
// Kernel 3: output projection, same 16x64-per-wave fp32 WMMA GEMM + bias.
// ---------------------------------------------------------------------------
__global__ void outproj_kernel(const float* __restrict__ A,
                               const float* __restrict__ W,
                               const float* __restrict__ bias,
                               float* __restrict__ out) {
    const int nb   = blockIdx.x;                 // 0..11
    const int wave = threadIdx.x >> 5;
    const int mt   = blockIdx.y * 4 + wave;      // 0..255
    const int lane = threadIdx.x & 31;
    const int col  = lane & 15;
    const int half = lane >> 4;
    const int m0   = mt * 16;
    const int nbase = nb * 64;

    const float* arow = A + (size_t)(m0 + col) * D_;
    v8f c0 = {}, c1 = {}, c2 = {}, c3 = {};
    for (int k = 0; k < D_; k += 4) {
        v2f a = *(const v2f*)(arow + k + half * 2);
        const float* w0 = W + (size_t)(k + half * 2) * D_ + nbase;
        const float* w1 = w0 + D_;
        v2f b;
        b.x = w0[col +  0]; b.y = w1[col +  0]; c0 = wmma_f32(a, b, c0);
        b.x = w0[col + 16]; b.y = w1[col + 16]; c1 = wmma_f32(a, b, c1);
        b.x = w0[col + 32]; b.y = w1[col + 32]; c2 = wmma_f32(a, b, c2);
        b.x = w0[col + 48]; b.y = w1[col + 48]; c3 = wmma_f32(a, b, c3);
    }
    v8f acc[4] = {c0, c1, c2, c3};
    for (int t = 0; t < 4; ++t) {
        const float bv = bias[nbase + t * 16 + col];
        for (int i = 0; i < 8; ++i) {
            const int m = m0 + i + half * 8;
            out[(size_t)m * D_ + nbase + t * 16 + col] = acc[t][i] + bv;
        }
    }
}

extern "C" void kernel_launch(void* const* d_in, const int* in_sizes, int n_in,
                              void* d_out, int out_size, void* d_ws, size_t ws_size,
                              hipStream_t stream) {
    (void)in_sizes; (void)n_in; (void)out_size; (void)ws_size;
    const float* X    = (const float*)d_in[0];
    const float* Wqkv = (const float*)d_in[1];
    const float* bqkv = (const float*)d_in[2];
    const float* Wout = (const float*)d_in[3];
    const float* bout = (const float*)d_in[4];

    float* out   = (float*)d_out;                        // [B,L,D]
    float* probs = out + (size_t)B_ * L_ * D_;           // [B,H,L,L]

    float* ws = (float*)d_ws;
    const size_t per = (size_t)B_ * H_ * L_ * HD_;       // 3,145,728 floats
    float* Q    = ws;
    float* K    = ws + per;
    float* V    = ws + 2 * per;
    float* attn = ws + 3 * per;                          // [B,L,D]

    // 1) QKV + bias + RoPE -> Q/K/V in [B,H,L,HD]
    qkv_rope_kernel<<<dim3(36, 64), 128, 0, stream>>>(X, Wqkv, bqkv, Q, K, V);

    // 2) fused causal attention; ~306KB dynamic LDS (<=320KB/WGP)
    const size_t smem =
        (size_t)(32 * L_ + 32 * HD_ + 32 * HD_ + 8 * 16 * KPAD_) * sizeof(float);
    hipFuncSetAttribute((const void*)attn_kernel,
                        hipFuncAttributeMaxDynamicSharedMemorySize, (int)smem);
    attn_kernel<<<dim3(L_ / 32, H_, B_), 256, smem, stream>>>(Q, K, V, attn, probs);

    // 3) output projection
    outproj_kernel<<<dim3(12, 64), 128, 0, stream>>>(attn, Wout, bout, out);
}